// iMMoE_Expert_Gate_58600533787422
// MI455X (gfx1250) — compile-verified
//
#include <hip/hip_runtime.h>
#include <hip/hip_bf16.h>

typedef float v2f __attribute__((ext_vector_type(2)));
typedef float v8f __attribute__((ext_vector_type(8)));

#define BDIM   4096
#define FDIM   2048
#define NEXP   8
#define NTA    4
#define NTASK  2
#define EDIM   128

#define BM 128
#define BN 64
#define BK 32
#define LDA2 (BM + 2)   // padded m-extent of k-pair A layout
#define LDB2 (BN + 2)   // padded n-extent of k-pair B layout

static __device__ __forceinline__ v8f wmma_f32(v2f a, v2f b, v8f c) {
    return __builtin_amdgcn_wmma_f32_16x16x4_f32(
        false, a, false, b, (short)0, c, false, false);
}

// Low 32 bits of a flat LDS pointer are the workgroup-relative LDS byte
// offset (ISA 10.2: LDS aperture lives entirely in addr[63:32]).
static __device__ __forceinline__ unsigned lds_off(const void* p) {
    return (unsigned)(uintptr_t)p;
}

static __device__ __forceinline__ void async_copy_b64(unsigned lds, const void* g) {
    asm volatile("global_load_async_to_lds_b64 %0, %1, off"
                 :: "v"(lds), "v"((unsigned long long)(uintptr_t)g) : "memory");
}
static __device__ __forceinline__ void async_copy_b32(unsigned lds, const void* g) {
    asm volatile("global_load_async_to_lds_b32 %0, %1, off"
                 :: "v"(lds), "v"((unsigned long long)(uintptr_t)g) : "memory");
}
static __device__ __forceinline__ void wait_async0() {
    asm volatile("s_wait_asynccnt 0x0" ::: "memory");
}

// ---------------------------------------------------------------------------
// Batched GEMM over 32 (expert x TA) pairs:
//   C[et, m, n] = sum_k A[et_or_bcast, m, k] * W[et, k, n] + bias[et, n]
// fp32 WMMA 16x16x4. Macro-tile 128x64, 8 waves (4 in M x 2 in N), each wave
// owns a 32x32 patch = 2x2 WMMA tiles.
// LDS tiles in k-pair layout [k/2][row][2] so each WMMA fragment is a single
// ds_load_b64. Tiles are moved by GLOBAL_LOAD_ASYNC_TO_LDS (ASYNCcnt),
// double-buffered, one barrier per K-step, copy fully overlapped with WMMAs.
// ---------------------------------------------------------------------------
__global__ __launch_bounds__(256)
void gemm_et_f32(const float* __restrict__ A, const float* __restrict__ W,
                 const float* __restrict__ bias, float* __restrict__ C,
                 int M, int N, int K, long long aBatchStride)
{
    const int et = blockIdx.z;
    const float* Ab = A + (long long)et * aBatchStride;
    const float* Wb = W + (long long)et * K * N;
    const float* bb = bias + (long long)et * N;
    float* Cb = C + (long long)et * M * N;

    const int m0 = blockIdx.y * BM;
    const int n0 = blockIdx.x * BN;

    __shared__ float Aps[2][BK / 2][LDA2][2];   // [k/2][m][k&1]
    __shared__ float Bps[2][BK / 2][LDB2][2];   // [k/2][n][k&1]

    const int tid   = threadIdx.x;
    const int lane  = tid & 31;
    const int wave  = tid >> 5;
    const int laneM = lane & 15;       // M/N position within fragment
    const int laneH = lane >> 4;       // K-pair select
    const int wm = (wave >> 1) * 32;   // wave row offset (0,32,64,96)
    const int wn = (wave & 1) * 32;    // wave col offset (0,32)

    const unsigned aLds[2] = { lds_off(&Aps[0][0][0][0]), lds_off(&Aps[1][0][0][0]) };
    const unsigned bLds[2] = { lds_off(&Bps[0][0][0][0]), lds_off(&Bps[1][0][0][0]) };

    const float* Abase = Ab + (long long)m0 * K;
    const float* Bbase = Wb + n0;

    // ---- async tile copy: whole block moves one (A,B) K-step tile --------
    // A: 128 rows x 32 k. Lane: kp = lane&15 (k-pair), mSub = lane>>4.
    //    Wave w copies rows w*16 + {0..15}, 2 rows per b64 instruction.
    // B: 32 k x 64 n. Wave w copies k rows w*4 + {0..3}, 32 n per b32 instr.
    const int kpL  = lane & 15;
    const int mSub = lane >> 4;

    auto issue_tiles = [&](int k0, int nb) {
        // A tile
        const float* ag = Abase + (long long)(wave * 16 + mSub) * K + k0 + 2 * kpL;
        unsigned al = aLds[nb] + (unsigned)((kpL * LDA2 + wave * 16 + mSub) * 8);
#pragma unroll
        for (int j = 0; j < 8; ++j) {
            async_copy_b64(al + (unsigned)(2 * j * 8),
                           ag + (long long)(2 * j) * K);
        }
        // B tile
#pragma unroll
        for (int j = 0; j < 8; ++j) {
            const int kr   = wave * 4 + (j >> 1);
            const int nOff = (j & 1) * 32 + lane;
            const float* bg = Bbase + (long long)(k0 + kr) * N + nOff;
            unsigned bl = bLds[nb] +
                (unsigned)(((kr >> 1) * LDB2 + nOff) * 8 + (kr & 1) * 4);
            async_copy_b32(bl, bg);
        }
    };

    v8f acc[2][2];
    acc[0][0] = v8f{}; acc[0][1] = v8f{};
    acc[1][0] = v8f{}; acc[1][1] = v8f{};

    // prologue: tile 0 -> buffer 0
    issue_tiles(0, 0);
    wait_async0();
    __syncthreads();

    const int nSteps = K / BK;
    int buf = 0;
    for (int s = 0; s < nSteps; ++s) {
        const bool hasNext = (s + 1 < nSteps);
        if (hasNext) {
            // safe: the barrier ending the previous step guarantees no wave
            // still reads buf^1; async engine fills it while we compute.
            issue_tiles((s + 1) * BK, buf ^ 1);
        }

        // 8 substeps x 4 WMMA = 32 WMMA on current buffer
#pragma unroll
        for (int kk = 0; kk < BK; kk += 4) {
            const int kp = (kk >> 1) + laneH;
            const v2f a0  = *(const v2f*)&Aps[buf][kp][wm + laneM][0];
            const v2f a1  = *(const v2f*)&Aps[buf][kp][wm + 16 + laneM][0];
            const v2f bv0 = *(const v2f*)&Bps[buf][kp][wn + laneM][0];
            const v2f bv1 = *(const v2f*)&Bps[buf][kp][wn + 16 + laneM][0];
            acc[0][0] = wmma_f32(a0, bv0, acc[0][0]);
            acc[0][1] = wmma_f32(a0, bv1, acc[0][1]);
            acc[1][0] = wmma_f32(a1, bv0, acc[1][0]);
            acc[1][1] = wmma_f32(a1, bv1, acc[1][1]);
        }

        if (hasNext) {
            wait_async0();     // my async writes to buf^1 done
            __syncthreads();   // everyone's writes done + done reading buf
            buf ^= 1;
        }
    }

    // epilogue with fused bias; C VGPR i holds rows {i, i+8} per lane half
#pragma unroll
    for (int mi = 0; mi < 2; mi++) {
        const int gm = m0 + wm + 16 * mi + 8 * laneH;
#pragma unroll
        for (int nj = 0; nj < 2; nj++) {
            const int cc = n0 + wn + 16 * nj + laneM;
            const float bv = bb[cc];
#pragma unroll
            for (int i = 0; i < 8; i++)
                Cb[(long long)(gm + i) * N + cc] = acc[mi][nj][i] + bv;
        }
    }
}

// ---------------------------------------------------------------------------
// E[e, b, d] = sum_t h3[e*4 + t, b, d]
// ---------------------------------------------------------------------------
__global__ __launch_bounds__(256)
void ta_sum_kernel(const float* __restrict__ h3, float* __restrict__ E)
{
    const long long PR = (long long)BDIM * EDIM;
    long long idx = (long long)blockIdx.x * 256 + threadIdx.x;
    long long e = idx / PR;
    long long r = idx - e * PR;
    const float* base = h3 + (e * NTA) * PR;
    E[idx] = base[r] + base[PR + r] + base[2 * PR + r] + base[3 * PR + r];
}

// ---------------------------------------------------------------------------
// g[t, b, e] = softmax_e( x[b,:] . gW[t,:,e] + gb[t,e] ); one block per (b,t)
// ---------------------------------------------------------------------------
__global__ __launch_bounds__(256)
void gate_softmax_kernel(const float* __restrict__ x, const float* __restrict__ gW,
                         const float* __restrict__ gb, float* __restrict__ g)
{
    const int b = blockIdx.x;
    const int t = blockIdx.y;
    const int tid = threadIdx.x;

    const float* xr  = x  + (long long)b * FDIM;
    const float* gwt = gW + (long long)t * FDIM * NEXP;

    float acc[NEXP];
#pragma unroll
    for (int e = 0; e < NEXP; e++) acc[e] = 0.0f;

    for (int f = tid; f < FDIM; f += 256) {
        const float xv = xr[f];
        const float4* gv = (const float4*)(gwt + (long long)f * NEXP);
        float4 g0 = gv[0];
        float4 g1 = gv[1];
        acc[0] += xv * g0.x; acc[1] += xv * g0.y;
        acc[2] += xv * g0.z; acc[3] += xv * g0.w;
        acc[4] += xv * g1.x; acc[5] += xv * g1.y;
        acc[6] += xv * g1.z; acc[7] += xv * g1.w;
    }

    __shared__ float sd[256];
    __shared__ float logit[NEXP];
#pragma unroll
    for (int e = 0; e < NEXP; e++) {
        sd[tid] = acc[e];
        __syncthreads();
        for (int s = 128; s > 0; s >>= 1) {
            if (tid < s) sd[tid] += sd[tid + s];
            __syncthreads();
        }
        if (tid == 0) logit[e] = sd[0] + gb[t * NEXP + e];
        __syncthreads();
    }

    if (tid == 0) {
        float m = logit[0];
#pragma unroll
        for (int e = 1; e < NEXP; e++) m = fmaxf(m, logit[e]);
        float ex[NEXP];
        float s = 0.0f;
#pragma unroll
        for (int e = 0; e < NEXP; e++) { ex[e] = __expf(logit[e] - m); s += ex[e]; }
        const float inv = 1.0f / s;
        float* go = g + ((long long)t * BDIM + b) * NEXP;
#pragma unroll
        for (int e = 0; e < NEXP; e++) go[e] = ex[e] * inv;
    }
}

// ---------------------------------------------------------------------------
// towers[t, b, d] = sum_e g[t, b, e] * E[e, b, d]; one block per (b,t)
// ---------------------------------------------------------------------------
__global__ __launch_bounds__(128)
void combine_kernel(const float* __restrict__ g, const float* __restrict__ E,
                    float* __restrict__ out)
{
    const int b = blockIdx.x;
    const int t = blockIdx.y;
    const int d = threadIdx.x;
    const float* gr = g + ((long long)t * BDIM + b) * NEXP;
    float s = 0.0f;
#pragma unroll
    for (int e = 0; e < NEXP; e++)
        s += gr[e] * E[((long long)e * BDIM + b) * EDIM + d];
    out[((long long)t * BDIM + b) * EDIM + d] = s;
}

// ---------------------------------------------------------------------------
extern "C" void kernel_launch(void* const* d_in, const int* in_sizes, int n_in,
                              void* d_out, int out_size, void* d_ws, size_t ws_size,
                              hipStream_t stream)
{
    const float* x  = (const float*)d_in[0];
    const float* W0 = (const float*)d_in[1];
    const float* b0 = (const float*)d_in[2];
    const float* W1 = (const float*)d_in[3];
    const float* b1 = (const float*)d_in[4];
    const float* W2 = (const float*)d_in[5];
    const float* b2 = (const float*)d_in[6];
    const float* W3 = (const float*)d_in[7];
    const float* b3 = (const float*)d_in[8];
    const float* gW = (const float*)d_in[9];
    const float* gb = (const float*)d_in[10];

    // workspace layout (bytes)
    const size_t H0_OFF = 0;                                   // 512 MiB
    const size_t H1_OFF = (size_t)32 * BDIM * 1024 * 4;        // 256 MiB
    const size_t E_OFF  = H1_OFF + (size_t)32 * BDIM * 512 * 4;
    const size_t G_OFF  = E_OFF + (size_t)NEXP * BDIM * EDIM * 4;

    float* h0 = (float*)((char*)d_ws + H0_OFF);
    float* h1 = (float*)((char*)d_ws + H1_OFF);
    float* h2 = h0;   // reuse: h0 dead once h1 is built
    float* h3 = h1;   // reuse: h1 dead once h2 is built
    float* Ebuf = (float*)((char*)d_ws + E_OFF);
    float* gbuf = (float*)((char*)d_ws + G_OFF);

    gemm_et_f32<<<dim3(1024 / BN, BDIM / BM, 32), 256, 0, stream>>>(
        x, W0, b0, h0, BDIM, 1024, 2048, 0LL);
    gemm_et_f32<<<dim3(512 / BN, BDIM / BM, 32), 256, 0, stream>>>(
        h0, W1, b1, h1, BDIM, 512, 1024, (long long)BDIM * 1024);
    gemm_et_f32<<<dim3(256 / BN, BDIM / BM, 32), 256, 0, stream>>>(
        h1, W2, b2, h2, BDIM, 256, 512, (long long)BDIM * 512);
    gemm_et_f32<<<dim3(128 / BN, BDIM / BM, 32), 256, 0, stream>>>(
        h2, W3, b3, h3, BDIM, 128, 256, (long long)BDIM * 256);

    {
        const long long total = (long long)NEXP * BDIM * EDIM;
        ta_sum_kernel<<<(unsigned)(total / 256), 256, 0, stream>>>(h3, Ebuf);
    }

    gate_softmax_kernel<<<dim3(BDIM, NTASK), 256, 0, stream>>>(x, gW, gb, gbuf);
    combine_kernel<<<dim3(BDIM, NTASK), 128, 0, stream>>>(gbuf, Ebuf, (float*)d_out);
}